// SpectralAttention_73212012527749
// MI455X (gfx1250) — compile-verified
//
#include <hip/hip_runtime.h>
#include <math.h>

// SpectralAttention for MI455X (gfx1250, wave32).
// Sizes fixed by the harness: B=1, L=1024, D=1024, H=16, HD=64, K=8.
#define SA_L   1024
#define SA_D   1024
#define SA_H   16
#define SA_HD  64
#define SA_K   8
#define SA_KD  (SA_D * SA_K)   // 8192
#define SA_EPS 1e-5f

typedef __attribute__((ext_vector_type(2))) float v2f;
typedef __attribute__((ext_vector_type(8))) float v8f;
typedef int sa_v4i __attribute__((__vector_size__(4 * sizeof(int))));

#ifndef __has_builtin
#define __has_builtin(x) 0
#endif
#if __has_builtin(__builtin_amdgcn_global_load_async_to_lds_b128)
#define SA_ASYNC 1
#else
#define SA_ASYNC 0
#endif

// 16-byte global -> LDS copy.  Async path: GLOBAL_LOAD_ASYNC_TO_LDS_B128
// (no VGPR round trip, tracked by ASYNCcnt).  Low 32 bits of a generic
// shared-memory address are the LDS byte offset (CDNA5 aperture rule).
__device__ __forceinline__ void sa_cp_f4(const float* __restrict__ g,
                                         float* __restrict__ l) {
#if SA_ASYNC
  __builtin_amdgcn_global_load_async_to_lds_b128(
      (__attribute__((address_space(1))) sa_v4i*)(unsigned long long)g,
      (__attribute__((address_space(3))) sa_v4i*)(unsigned int)(unsigned long long)l,
      0, 0);
#else
  *(float4*)l = *(const float4*)g;
#endif
}

__device__ __forceinline__ void sa_wait_async_prev_stage(bool have_next) {
#if SA_ASYNC
  if (have_next)
    asm volatile("s_wait_asynccnt 0x8" ::: "memory");   // next stage (8 loads) still in flight
  else
    asm volatile("s_wait_asynccnt 0x0" ::: "memory");
#else
  (void)have_next;
#endif
}

// ---------------------------------------------------------------------------
// K1: causal depthwise-expand conv (kernel 3, pad (2,0)) + (H,h,K)->(K,H,h)
// feature reorder.  xr[l][k*D + d] = conv_out[l][d*K + k] + conv_b[d*K+k]
// ---------------------------------------------------------------------------
__global__ __launch_bounds__(256) void sa_conv_reorder(
    const float* __restrict__ x, const float* __restrict__ cw,
    const float* __restrict__ cb, float* __restrict__ xr) {
  int idx = blockIdx.x * blockDim.x + threadIdx.x;   // over L*D
  if (idx >= SA_L * SA_D) return;
  int l = idx / SA_D;
  int d = idx - l * SA_D;
  float x0 = (l >= 2) ? x[(l - 2) * SA_D + d] : 0.f;
  float x1 = (l >= 1) ? x[(l - 1) * SA_D + d] : 0.f;
  float x2 = x[l * SA_D + d];
#pragma unroll
  for (int k = 0; k < SA_K; ++k) {
    int ch = d * SA_K + k;
    float r = cw[ch * 3 + 0] * x0 + cw[ch * 3 + 1] * x1 + cw[ch * 3 + 2] * x2 + cb[ch];
    xr[(size_t)l * SA_KD + k * SA_D + d] = r;
  }
}

// ---------------------------------------------------------------------------
// K2: fp32 WMMA GEMM  C(MxN) = A(MxK) @ W(KxN) + bias
// 64x64 tile / 256-thread block. K staged through double-buffered LDS in
// 64-chunks via async LDS copies; V_WMMA_F32_16X16X4_F32; 8 waves, each wave
// owns one 16-row x 32-col strip (2 accumulators).  Dims multiples of 64.
// ---------------------------------------------------------------------------
__global__ __launch_bounds__(256) void sa_gemm_wmma_f32(
    const float* __restrict__ A, const float* __restrict__ W,
    const float* __restrict__ bias, float* __restrict__ C,
    int M, int N, int Kd) {
  __shared__ float As[2][64 * 64];
  __shared__ float Ws[2][64 * 64];
  const int tid = threadIdx.x;
  const int m0 = blockIdx.y * 64;
  const int n0 = blockIdx.x * 64;
  const int wv = tid >> 5;           // wave id 0..7
  const int lane = tid & 31;
  const int rT = wv & 3;             // 16-row strip within the 64-row tile
  const int cP = wv >> 2;            // 32-col half within the 64-col tile
  const int lm = lane & 15;          // lane row/col within 16
  const int lh = lane >> 4;          // lane half
  // cooperative load mapping: 16 rows x 16 float4 per pass, 4 passes
  const int r = tid >> 4;            // 0..15
  const int c4 = (tid & 15) << 2;    // 0,4,...,60

  v8f acc0 = {};
  v8f acc1 = {};

  const int nStages = Kd >> 6;

  // prologue: stage 0 into buffer 0  (8 x b128 per thread)
#pragma unroll
  for (int rr = 0; rr < 64; rr += 16) {
    sa_cp_f4(A + (size_t)(m0 + r + rr) * Kd + c4, &As[0][(r + rr) * 64 + c4]);
    sa_cp_f4(W + (size_t)(r + rr) * N + n0 + c4, &Ws[0][(r + rr) * 64 + c4]);
  }

  for (int s = 0; s < nStages; ++s) {
    const int cur = s & 1;
    const bool have_next = (s + 1 < nStages);
    if (have_next) {
      const int kb = (s + 1) << 6;
#pragma unroll
      for (int rr = 0; rr < 64; rr += 16) {
        sa_cp_f4(A + (size_t)(m0 + r + rr) * Kd + kb + c4,
                 &As[cur ^ 1][(r + rr) * 64 + c4]);
        sa_cp_f4(W + (size_t)(kb + r + rr) * N + n0 + c4,
                 &Ws[cur ^ 1][(r + rr) * 64 + c4]);
      }
    }
    sa_wait_async_prev_stage(have_next);
    __syncthreads();

#pragma unroll
    for (int kk = 0; kk < 64; kk += 4) {
      const int ka = kk + (lh << 1);
      v2f a;
      a.x = As[cur][(rT * 16 + lm) * 64 + ka];
      a.y = As[cur][(rT * 16 + lm) * 64 + ka + 1];
      v2f b0;
      b0.x = Ws[cur][ka * 64 + cP * 32 + lm];
      b0.y = Ws[cur][(ka + 1) * 64 + cP * 32 + lm];
      v2f b1;
      b1.x = Ws[cur][ka * 64 + cP * 32 + 16 + lm];
      b1.y = Ws[cur][(ka + 1) * 64 + cP * 32 + 16 + lm];
      acc0 = __builtin_amdgcn_wmma_f32_16x16x4_f32(false, a, false, b0,
                                                   (short)0, acc0, false, false);
      acc1 = __builtin_amdgcn_wmma_f32_16x16x4_f32(false, a, false, b1,
                                                   (short)0, acc1, false, false);
    }
    __syncthreads();
  }

  const int nc0 = n0 + cP * 32 + lm;
  const int nc1 = nc0 + 16;
  const float bb0 = bias ? bias[nc0] : 0.f;
  const float bb1 = bias ? bias[nc1] : 0.f;
#pragma unroll
  for (int rr8 = 0; rr8 < 8; ++rr8) {
    const int row = m0 + rT * 16 + rr8 + (lh << 3);
    C[(size_t)row * N + nc0] = acc0[rr8] + bb0;
    C[(size_t)row * N + nc1] = acc1[rr8] + bb1;
  }
}

// ---------------------------------------------------------------------------
// K3: per-head L2 normalize + relayout  (L,D) -> (H,L,64)
// one 64-thread block per (l,h)
// ---------------------------------------------------------------------------
__global__ __launch_bounds__(64) void sa_norm_heads(
    const float* __restrict__ src, float* __restrict__ dst) {
  __shared__ float red[64];
  const int b = blockIdx.x;
  const int l = b >> 4;
  const int h = b & 15;
  const int tid = threadIdx.x;
  float v = src[(size_t)l * SA_D + h * SA_HD + tid];
  red[tid] = v * v;
  __syncthreads();
  for (int ofs = 32; ofs >= 1; ofs >>= 1) {
    if (tid < ofs) red[tid] += red[tid + ofs];
    __syncthreads();
  }
  float nrm = fmaxf(sqrtf(red[0]), 1e-12f);
  dst[((size_t)h * SA_L + l) * SA_HD + tid] = v / nrm;
}

// ---------------------------------------------------------------------------
// K4: per-head sequential scan.  One block per head, 256 threads.
// State in LDS: Z (64x64), nstate (64), scalars m/s/g_c.
// Per position: g = leakyrelu(k^T (kv_scale.*wgz) v + b)+eps;
//   Z += g * kv_scale .* (k (x) v);  (m,s,n) logsumexp update;
//   Y = Ybase + (qsum*lin_attn - Ybase) * w
// ---------------------------------------------------------------------------
__global__ __launch_bounds__(256) void sa_scan(
    const float* __restrict__ qn, const float* __restrict__ kn,
    const float* __restrict__ vn, const float* __restrict__ kv_scale,
    const float* __restrict__ wgz_w, const float* __restrict__ wgz_b,
    const float* __restrict__ qk_scale, float* __restrict__ Y) {
  __shared__ float Zs[4096];
  __shared__ float Ms[4096];    // kv_scale .* wgz_w
  __shared__ float KVs[4096];   // kv_scale
  __shared__ float qv[64], kvv[64], vv[64];
  __shared__ float red[256];
  __shared__ float ybp[4][64];
  __shared__ float nst[64];
  __shared__ float sc[10];      // 0:m 1:s 2:gc 3:ex 4:et 5:g 6:1/(s+e) 7:1/(gc+e) 8:qsum

  const int h = blockIdx.x;
  const int tid = threadIdx.x;

  for (int i = tid; i < 4096; i += 256) {
    float kvsv = kv_scale[i];
    KVs[i] = kvsv;
    Ms[i] = kvsv * wgz_w[i];
    Zs[i] = 0.f;
  }
  if (tid < 64) nst[tid] = 0.f;
  if (tid == 0) { sc[0] = -3.4e38f; sc[1] = 0.f; sc[2] = 0.f; }
  __syncthreads();

  const float qks = qk_scale[h];
  const float gzb = wgz_b[0];
  const float* qh = qn + (size_t)h * SA_L * SA_HD;
  const float* kh = kn + (size_t)h * SA_L * SA_HD;
  const float* vh = vn + (size_t)h * SA_L * SA_HD;

  const int p = tid >> 2;            // row 0..63 (bilinear / Z update)
  const int nb = (tid & 3) << 4;     // 16-col slab
  const int nn = tid & 63;           // col (ybase)
  const int pr = tid >> 6;           // row group 0..3 (ybase)

  for (int t = 0; t < SA_L; ++t) {
    if (tid < 64) {
      qv[tid] = qh[t * 64 + tid];
      kvv[tid] = kh[t * 64 + tid];
      vv[tid] = vh[t * 64 + tid];
    }
    __syncthreads();

    // bilinear form k^T M v  (256 partials)
    const float kp = kvv[p];
    float part = 0.f;
#pragma unroll
    for (int j = 0; j < 16; ++j) part += Ms[p * 64 + nb + j] * vv[nb + j];
    red[tid] = part * kp;
    __syncthreads();
    for (int ofs = 128; ofs >= 1; ofs >>= 1) {
      if (tid < ofs) red[tid] += red[tid + ofs];
      __syncthreads();
    }
    const float bil = red[0];
    __syncthreads();

    // q.k dot and sum(q): two 64-wide reductions side by side
    if (tid < 64) {
      red[tid] = qv[tid] * kvv[tid];
      red[tid + 64] = qv[tid];
    }
    __syncthreads();
    for (int ofs = 32; ofs >= 1; ofs >>= 1) {
      if (tid < ofs) red[tid] += red[tid + ofs];
      else if (tid >= 64 && tid < 64 + ofs) red[tid] += red[tid + ofs];
      __syncthreads();
    }

    if (tid == 0) {
      float zg = bil + gzb;
      float g = (zg >= 0.f ? zg : 0.01f * zg) + SA_EPS;  // leaky_relu + eps
      float sim = red[0] * qks;
      float m_old = sc[0];
      float m_new = fmaxf(m_old, sim);
      float ex = expf(m_old - m_new);
      float et = expf(sim - m_new);
      float s_new = sc[1] * ex + et;
      float gc = sc[2] + g;
      sc[0] = m_new; sc[1] = s_new; sc[2] = gc;
      sc[3] = ex; sc[4] = et; sc[5] = g;
      sc[6] = 1.f / (s_new + SA_EPS);
      sc[7] = 1.f / (gc + SA_EPS);
      sc[8] = red[64];                 // qsum
    }
    __syncthreads();

    // Z state update (16 entries per thread) and n state update
    const float g = sc[5];
#pragma unroll
    for (int j = 0; j < 16; ++j) {
      const int idx = p * 64 + nb + j;
      Zs[idx] += g * KVs[idx] * kp * vv[nb + j];
    }
    if (tid < 64) nst[tid] = nst[tid] * sc[3] + sc[4] * vv[tid];
    __syncthreads();

    // Ybase = (q^T Z)/(gc+eps): 4 partial rows of 16
    float yp = 0.f;
#pragma unroll
    for (int i = 0; i < 16; ++i) {
      const int pp = pr * 16 + i;
      yp += qv[pp] * Zs[pp * 64 + nn];
    }
    ybp[pr][nn] = yp;
    __syncthreads();

    if (tid < 64) {
      float yb = (ybp[0][tid] + ybp[1][tid] + ybp[2][tid] + ybp[3][tid]) * sc[7];
      float w = sc[4] * sc[6];                // exp(sim-m)/(s+eps)
      float ylin = sc[8] * nst[tid] * sc[6];  // qsum * n/(s+eps)
      Y[(size_t)t * SA_D + h * SA_HD + tid] = yb + (ylin - yb) * w;
    }
    __syncthreads();
  }
}

// ---------------------------------------------------------------------------
// K5: in-place row L2 normalize over D
// ---------------------------------------------------------------------------
__global__ __launch_bounds__(256) void sa_norm_rows(float* __restrict__ Y) {
  __shared__ float red[256];
  const int l = blockIdx.x;
  const int tid = threadIdx.x;
  float ss = 0.f;
  for (int d = tid; d < SA_D; d += 256) {
    float v = Y[(size_t)l * SA_D + d];
    ss += v * v;
  }
  red[tid] = ss;
  __syncthreads();
  for (int ofs = 128; ofs >= 1; ofs >>= 1) {
    if (tid < ofs) red[tid] += red[tid + ofs];
    __syncthreads();
  }
  float inv = 1.f / fmaxf(sqrtf(red[0]), 1e-12f);
  for (int d = tid; d < SA_D; d += 256) Y[(size_t)l * SA_D + d] *= inv;
}

// ---------------------------------------------------------------------------
extern "C" void kernel_launch(void* const* d_in, const int* in_sizes, int n_in,
                              void* d_out, int out_size, void* d_ws, size_t ws_size,
                              hipStream_t stream) {
  (void)in_sizes; (void)n_in; (void)out_size; (void)ws_size;
  const float* x      = (const float*)d_in[0];
  const float* conv_w = (const float*)d_in[1];
  const float* conv_b = (const float*)d_in[2];
  const float* mphi_w = (const float*)d_in[3];
  const float* mphi_b = (const float*)d_in[4];
  const float* wq_w   = (const float*)d_in[5];
  const float* wq_b   = (const float*)d_in[6];
  const float* wk_w   = (const float*)d_in[7];
  const float* wk_b   = (const float*)d_in[8];
  const float* wv_w   = (const float*)d_in[9];
  const float* wv_b   = (const float*)d_in[10];
  const float* wo_w   = (const float*)d_in[11];
  const float* wo_b   = (const float*)d_in[12];
  const float* wgz_w  = (const float*)d_in[13];
  const float* wgz_b  = (const float*)d_in[14];
  const float* kv_sc  = (const float*)d_in[15];
  const float* qk_sc  = (const float*)d_in[16];

  // workspace layout (floats): xr (L*8192) | xt (L*D) | Y (L*D)  = 40 MB total.
  // xr region is dead after the mphi GEMM and is reused for q/k/v buffers.
  float* ws = (float*)d_ws;
  float* xr = ws;
  float* xt = ws + (size_t)SA_L * SA_KD;
  float* Y  = xt + (size_t)SA_L * SA_D;
  float* qraw = xr;
  float* kraw = xr + 1 * (size_t)SA_L * SA_D;
  float* vraw = xr + 2 * (size_t)SA_L * SA_D;
  float* qn_  = xr + 3 * (size_t)SA_L * SA_D;
  float* kn_  = xr + 4 * (size_t)SA_L * SA_D;
  float* vn_  = xr + 5 * (size_t)SA_L * SA_D;

  sa_conv_reorder<<<(SA_L * SA_D + 255) / 256, 256, 0, stream>>>(x, conv_w, conv_b, xr);

  dim3 g16(SA_D / 64, SA_L / 64);
  sa_gemm_wmma_f32<<<g16, 256, 0, stream>>>(xr, mphi_w, mphi_b, xt, SA_L, SA_D, SA_KD);
  sa_gemm_wmma_f32<<<g16, 256, 0, stream>>>(xt, wq_w, wq_b, qraw, SA_L, SA_D, SA_D);
  sa_gemm_wmma_f32<<<g16, 256, 0, stream>>>(xt, wk_w, wk_b, kraw, SA_L, SA_D, SA_D);
  sa_gemm_wmma_f32<<<g16, 256, 0, stream>>>(xt, wv_w, wv_b, vraw, SA_L, SA_D, SA_D);

  sa_norm_heads<<<SA_L * SA_H, 64, 0, stream>>>(qraw, qn_);
  sa_norm_heads<<<SA_L * SA_H, 64, 0, stream>>>(kraw, kn_);
  sa_norm_heads<<<SA_L * SA_H, 64, 0, stream>>>(vraw, vn_);

  sa_scan<<<SA_H, 256, 0, stream>>>(qn_, kn_, vn_, kv_sc, wgz_w, wgz_b, qk_sc, Y);

  sa_norm_rows<<<SA_L, 256, 0, stream>>>(Y);

  sa_gemm_wmma_f32<<<g16, 256, 0, stream>>>(Y, wo_w, wo_b, (float*)d_out, SA_L, SA_D, SA_D);
}